// QIncMeasurement_38276748541964
// MI455X (gfx1250) — compile-verified
//
#include <hip/hip_runtime.h>
#include <stdint.h>

typedef float v2f __attribute__((ext_vector_type(2)));
typedef float v4f __attribute__((ext_vector_type(4)));
typedef float v8f __attribute__((ext_vector_type(8)));

#define DIM 64
#define RHO_LD 72   // LDS dword stride for rho rows (banks (8k+n)%64: halves disjoint)
#define X_LD 68     // LDS dword stride for x tile rows (banks (4m+c)%64)
#define WPB 8       // waves per block (256 threads, wave32)
#define BLOCK 256

// ds_swizzle_b32 group-of-32 XOR pattern: offset = {1'b0, xor[4:0], or[4:0], and[4:0]}
#define XOR_PAT(m) (((m) << 10) | 0x1F)

template <int PAT>
__device__ __forceinline__ float swz(float v) {
  return __int_as_float(__builtin_amdgcn_ds_swizzle(__float_as_int(v), PAT));
}

__global__ __launch_bounds__(BLOCK) void qmeas_wmma_f32(
    const float* __restrict__ x,
    const float* __restrict__ rho,
    float* __restrict__ out,
    int numTiles)
{
  __shared__ float sRho[DIM * RHO_LD];      // 18432 B
  __shared__ float sX[WPB][2][16 * X_LD];   // 69632 B, double-buffered per wave

  const int tid  = threadIdx.x;
  const int lane = tid & 31;
  const int wib  = tid >> 5;
  const int hl   = (lane >> 4) & 1;   // lane half (K/M high bit in WMMA layouts)
  const int li   = lane & 15;

  // Stage rho (64x64 row-major) into padded LDS: 1024 float4s over 256 threads.
  {
    const v4f* r4 = (const v4f*)rho;
    #pragma unroll
    for (int it = 0; it < 4; ++it) {
      int i   = tid + it * BLOCK;
      int row = i >> 4;
      int c4  = i & 15;
      v4f v = r4[i];
      *(v4f*)&sRho[row * RHO_LD + c4 * 4] = v;
    }
  }
  __syncthreads();

  // Preload ALL rho B-fragments into registers once (128 VGPRs); survives the
  // asm "memory" clobbers in the tile loop. B fragment (16x16x4 f32): N=li
  // striped across lanes, VGPR j holds K = 4kc + 2hl + j.
  v2f Bfrag[4][16];
  #pragma unroll
  for (int nt = 0; nt < 4; ++nt) {
    int n = nt * 16 + li;
    #pragma unroll
    for (int kc = 0; kc < 16; ++kc) {
      int k0 = 4 * kc + 2 * hl;
      v2f b;
      b.x = sRho[k0 * RHO_LD + n];
      b.y = sRho[(k0 + 1) * RHO_LD + n];
      Bfrag[nt][kc] = b;
    }
  }

  // Scalarized tile loop (wave-uniform in SGPRs).
  const int gw = __builtin_amdgcn_readfirstlane((int)(blockIdx.x * WPB + wib));
  const int ws = (int)(gridDim.x * WPB);

  // Async staging geometry: float4 id f = lane + 32q.
  //   global byte off = lane*16 + q*512 ; LDS byte off = base + q*544
  const unsigned ldsBuf0 =
      (unsigned)(uintptr_t)&sX[wib][0][(lane >> 4) * X_LD + (lane & 15) * 4];
  const unsigned ldsBuf1 =
      (unsigned)(uintptr_t)&sX[wib][1][(lane >> 4) * X_LD + (lane & 15) * 4];
  const uint64_t xlane = (uint64_t)(uintptr_t)x + (unsigned)(lane * 16);

  // Issue the 8x B128 async DMA staging a 16x64 tile into one LDS buffer.
  auto stage = [&](unsigned ldsBase, int tile) {
    uint64_t g = xlane + ((uint64_t)(unsigned)tile << 12);   // tile*4096 B
    #pragma unroll
    for (int q = 0; q < 8; ++q) {
      unsigned loff = ldsBase + (unsigned)(q * 544);
      uint64_t ga   = g + (unsigned)(q * 512);
      asm volatile("global_load_async_to_lds_b128 %0, %1, off"
                   :: "v"(loff), "v"(ga) : "memory");
    }
  };

  auto compute = [&](const float* mx, int tile) {
    // A fragments (16x4 f32): lane holds row li; VGPR j = K=(4kc+2hl+j).
    v2f A[16];
    #pragma unroll
    for (int kc = 0; kc < 16; ++kc) {
      int k0 = 4 * kc + 2 * hl;
      v2f a;
      a.x = mx[li * X_LD + k0];
      a.y = mx[li * X_LD + k0 + 1];
      A[kc] = a;
    }

    // norm2 from A fragments; one cross-half combine -> lane li has norm2(row li).
    float npart = 0.f;
    #pragma unroll
    for (int kc = 0; kc < 16; ++kc) {
      npart = __builtin_fmaf(A[kc].x, A[kc].x, npart);
      npart = __builtin_fmaf(A[kc].y, A[kc].y, npart);
    }
    npart += swz<XOR_PAT(16)>(npart);

    float qp[8];
    #pragma unroll
    for (int r = 0; r < 8; ++r) qp[r] = 0.f;

    // Y = Xtile @ rho (f32 WMMA), fused with the quad dot.
    #pragma unroll
    for (int nt = 0; nt < 4; ++nt) {
      v8f acc = {0.f, 0.f, 0.f, 0.f, 0.f, 0.f, 0.f, 0.f};
      int n = nt * 16 + li;
      #pragma unroll
      for (int kc = 0; kc < 16; ++kc) {
        acc = __builtin_amdgcn_wmma_f32_16x16x4_f32(
            false, A[kc], false, Bfrag[nt][kc], (short)0, acc, false, false);
      }
      #pragma unroll
      for (int r = 0; r < 8; ++r) {           // D: VGPR r <-> row r + 8*hl
        float xv = mx[(r + 8 * hl) * X_LD + n];
        qp[r] = __builtin_fmaf(xv, acc[r], qp[r]);
      }
    }

    // Reduce-scatter butterfly over the 16 column-lanes using ds_swizzle XOR
    // patterns: keep the register half selected by the li bit, ship the other;
    // ends with quad(row li) in lane li of each half (rows 0..7 lo, 8..15 hi).
    const bool b4 = (li & 4) != 0;
    float t4[4];
    #pragma unroll
    for (int j = 0; j < 4; ++j) {
      float keep = b4 ? qp[j + 4] : qp[j];
      float send = b4 ? qp[j] : qp[j + 4];
      t4[j] = keep + swz<XOR_PAT(4)>(send);
    }
    const bool b2 = (li & 2) != 0;
    float t2v[2];
    #pragma unroll
    for (int j = 0; j < 2; ++j) {
      float keep = b2 ? t4[j + 2] : t4[j];
      float send = b2 ? t4[j] : t4[j + 2];
      t2v[j] = keep + swz<XOR_PAT(2)>(send);
    }
    const bool b1 = (li & 1) != 0;
    float keep = b1 ? t2v[1] : t2v[0];
    float send = b1 ? t2v[0] : t2v[1];
    float qv = keep + swz<XOR_PAT(1)>(send);
    qv += swz<XOR_PAT(8)>(qv);                // plain step completes 16-lane sum

    bool active = hl ? (li >= 8) : (li < 8);
    if (active) out[(size_t)tile * 16 + li] = npart * qv;
  };

  // Double-buffered pipeline: prefetch t+1 while computing t. Async loads
  // complete in order, so asynccnt<=8 after issuing the next batch guarantees
  // the current buffer's 8-op batch has landed.
  int tile = gw;
  if (tile < numTiles) stage(ldsBuf0, tile);
  for (; tile < numTiles; tile += 2 * ws) {
    {
      int nxt = tile + ws;
      if (nxt < numTiles) {
        asm volatile("s_wait_dscnt 0x0" ::: "memory");   // drain reads of buf1
        stage(ldsBuf1, nxt);
        asm volatile("s_wait_asynccnt 0x8" ::: "memory");
      } else {
        asm volatile("s_wait_asynccnt 0x0" ::: "memory");
      }
      compute(&sX[wib][0][0], tile);
    }
    int t2 = tile + ws;
    if (t2 < numTiles) {
      int nxt = t2 + ws;
      if (nxt < numTiles) {
        asm volatile("s_wait_dscnt 0x0" ::: "memory");   // drain reads of buf0
        stage(ldsBuf0, nxt);
        asm volatile("s_wait_asynccnt 0x8" ::: "memory");
      } else {
        asm volatile("s_wait_asynccnt 0x0" ::: "memory");
      }
      compute(&sX[wib][1][0], t2);
    }
  }
}

extern "C" void kernel_launch(void* const* d_in, const int* in_sizes, int n_in,
                              void* d_out, int out_size, void* d_ws, size_t ws_size,
                              hipStream_t stream) {
  const float* x   = (const float*)d_in[0];
  const float* rho = (const float*)d_in[1];
  float* out = (float*)d_out;

  int batch    = in_sizes[0] / DIM;   // 4194304
  int numTiles = batch >> 4;          // 262144 row-tiles of 16

  int grid = 4096;                    // 32768 waves -> 8 tiles/wave grid-stride
  if (grid * WPB > numTiles) grid = (numTiles + WPB - 1) / WPB;
  if (grid < 1) grid = 1;
  qmeas_wmma_f32<<<grid, BLOCK, 0, stream>>>(x, rho, out, numTiles);
}